// MambaLayer_14705968022120
// MI455X (gfx1250) — compile-verified
//
#include <hip/hip_runtime.h>
#include <math.h>

// ---------------- problem constants ----------------
constexpr int S_  = 4;     // SPLITS
constexpr int D_  = 32;    // channels per split
constexpr int DI_ = 64;    // expanded inner dim
constexpr int DS_ = 16;    // state dim
constexpr int L_  = 4096;  // H*W
constexpr int B_  = 4;     // batch
constexpr int C_  = 128;   // channels
constexpr int BL_ = B_ * L_;          // 16384 rows
constexpr int NP_ = 48;    // x_proj width padded 34 -> 48 (3 WMMA tiles)
constexpr float EPS_ = 1e-5f;

// workspace layout (float offsets)
constexpr size_t OFF_XNP  = 0;                                    // (S,BL,32)
constexpr size_t OFF_XZ   = OFF_XNP  + (size_t)S_*BL_*D_;         // (S,BL,128)
constexpr size_t OFF_XC   = OFF_XZ   + (size_t)S_*BL_*2*DI_;      // (S,BL,64)
constexpr size_t OFF_PR   = OFF_XC   + (size_t)S_*BL_*DI_;        // (S,BL,48)
constexpr size_t OFF_YG   = OFF_PR   + (size_t)S_*BL_*NP_;        // (S,BL,64)
constexpr size_t OFF_YC   = OFF_YG   + (size_t)S_*BL_*DI_;        // (BL,128)
constexpr size_t OFF_INWT = OFF_YC   + (size_t)BL_*C_;            // (S,128,32)
constexpr size_t OFF_XPWT = OFF_INWT + (size_t)S_*C_*D_;          // (S,48,64)
constexpr size_t OFF_OWT  = OFF_XPWT + (size_t)S_*NP_*DI_;        // (S,32,64)
constexpr size_t OFF_PWT  = OFF_OWT  + (size_t)S_*D_*DI_;         // (S,32,32)

// ---------------- WMMA helpers (fp32 16x16x4) ----------------
typedef __attribute__((ext_vector_type(2))) float v2f;
typedef __attribute__((ext_vector_type(8))) float v8f;

__device__ inline v8f wmma4(v2f a, v2f b, v8f c) {
  // 8 args: (neg_a, A, neg_b, B, c_mod, C, reuse_a, reuse_b)
  return __builtin_amdgcn_wmma_f32_16x16x4_f32(false, a, false, b, (short)0, c,
                                               false, false);
}

// C[16x16] += A[16xK](lda, row-major) * B[Kx16] given as BT[16xK](ldbt) --
// both operands K-contiguous so a/b pairs merge into global_load_b64.
__device__ inline void gemm_tile_t(const float* __restrict__ A, int lda,
                                   const float* __restrict__ BT, int ldbt,
                                   int K, v8f& acc) {
  const int lane = threadIdx.x & 31;
  const int hi = lane >> 4;
  const int lr = lane & 15;
  for (int k = 0; k < K; k += 4) {
    v2f a, b;
    a.x = A[lr * lda + k + 2 * hi];
    a.y = A[lr * lda + k + 2 * hi + 1];
    b.x = BT[lr * ldbt + k + 2 * hi];
    b.y = BT[lr * ldbt + k + 2 * hi + 1];
    acc = wmma4(a, b, acc);
  }
}

__device__ inline void store_tile(float* __restrict__ Cp, int ldc, v8f d) {
  const int lane = threadIdx.x & 31;
  const int hi = lane >> 4;
  const int lr = lane & 15;
#pragma unroll
  for (int r = 0; r < 8; ++r) Cp[(r + 8 * hi) * ldc + lr] = d[r];
}

__device__ inline float wave_sum(float v) {
#pragma unroll
  for (int m = 16; m >= 1; m >>= 1) v += __shfl_xor(v, m, 32);
  return v;
}

__device__ inline float silu_f(float x) { return x / (1.f + __expf(-x)); }

// ---------------- kernel 0: transpose/pad all weight matrices ----------------
__global__ __launch_bounds__(256) void prep_kernel(
    const float* __restrict__ in_w, const float* __restrict__ xpw,
    const float* __restrict__ ow, const float* __restrict__ pw,
    float* __restrict__ in_wT, float* __restrict__ xpwT,
    float* __restrict__ owT, float* __restrict__ pwT) {
  int t = blockIdx.x * 256 + threadIdx.x;
  if (t < S_ * C_ * D_) {  // in_wT (s, n=128, k=32)
    int k = t & 31, nk = t >> 5, n = nk & 127, s = nk >> 7;
    in_wT[t] = in_w[(s * D_ + k) * (2 * DI_) + n];
    return;
  }
  int t2 = t - S_ * C_ * D_;
  if (t2 < S_ * NP_ * DI_) {  // xpwT (s, n=48, k=64), pad n>=34 with 0
    int k = t2 & 63, nk = t2 >> 6, n = nk % NP_, s = nk / NP_;
    xpwT[t2] = (n < 2 + 2 * DS_) ? xpw[(s * DI_ + k) * (2 + 2 * DS_) + n] : 0.f;
    return;
  }
  int t3 = t2 - S_ * NP_ * DI_;
  if (t3 < S_ * D_ * DI_) {  // owT (s, n=32, k=64)
    int k = t3 & 63, nk = t3 >> 6, n = nk & 31, s = nk >> 5;
    owT[t3] = ow[(s * DI_ + k) * D_ + n];
    return;
  }
  int t4 = t3 - S_ * D_ * DI_;
  if (t4 < S_ * D_ * D_) {  // pwT (s, n=32, k=32)
    int k = t4 & 31, nk = t4 >> 5, n = nk & 31, s = nk >> 5;
    pwT[t4] = pw[(s * D_ + k) * D_ + n];
  }
}

// ---------------- kernel 1: double layernorm, NCHW -> (S,BL,32) ----------------
__global__ __launch_bounds__(256) void ln_kernel(
    const float* __restrict__ x, const float* __restrict__ g,
    const float* __restrict__ bta, const float* __restrict__ pg,
    const float* __restrict__ pb, float* __restrict__ xnp) {
  const int row = blockIdx.x * 8 + (threadIdx.x >> 5);  // p in [0, BL)
  const int lane = threadIdx.x & 31;
  const int b = row >> 12;
  const int l = row & (L_ - 1);
  float v[4];
#pragma unroll
  for (int j = 0; j < 4; ++j)
    v[j] = x[((size_t)b * C_ + lane + 32 * j) * L_ + l];
  float sum = wave_sum(v[0] + v[1] + v[2] + v[3]);
  const float mu = sum * (1.f / 128.f);
  float q = 0.f;
#pragma unroll
  for (int j = 0; j < 4; ++j) { float d = v[j] - mu; q += d * d; }
  q = wave_sum(q);
  const float inv = rsqrtf(q * (1.f / 128.f) + EPS_);
  float xn[4];
#pragma unroll
  for (int j = 0; j < 4; ++j) {
    const int c = lane + 32 * j;
    xn[j] = (v[j] - mu) * inv * g[c] + bta[c];
  }
  // per-split LN: split j lives exactly in xn[j] across the 32 lanes
#pragma unroll
  for (int j = 0; j < 4; ++j) {
    float m2 = wave_sum(xn[j]) * (1.f / 32.f);
    float d = xn[j] - m2;
    float q2 = wave_sum(d * d) * (1.f / 32.f);
    float o = d * rsqrtf(q2 + EPS_) * pg[j * D_ + lane] + pb[j * D_ + lane];
    xnp[((size_t)j * BL_ + row) * D_ + lane] = o;
  }
}

// ---------------- kernel 2: in_proj GEMM (BL,32)@(32,128) per split ----------------
__global__ __launch_bounds__(128) void inproj_kernel(
    const float* __restrict__ xnp, const float* __restrict__ in_wT,
    float* __restrict__ xz) {
  int widx = blockIdx.x * 4 + (threadIdx.x >> 5);  // S*1024*8 tiles
  int nt = widx & 7;
  int mt = (widx >> 3) & 1023;
  int s = widx >> 13;
  const float* A = xnp + ((size_t)s * BL_ + mt * 16) * D_;
  const float* BT = in_wT + (size_t)s * C_ * D_ + nt * 16 * D_;
  v8f acc = {0.f, 0.f, 0.f, 0.f, 0.f, 0.f, 0.f, 0.f};
  gemm_tile_t(A, D_, BT, D_, D_, acc);
  store_tile(xz + ((size_t)s * BL_ + mt * 16) * 2 * DI_ + nt * 16, 2 * DI_, acc);
}

// ---------------- kernel 3: causal depthwise conv + SiLU ----------------
__global__ __launch_bounds__(256) void conv_kernel(
    const float* __restrict__ xz, const float* __restrict__ cw,
    const float* __restrict__ cb, float* __restrict__ xc) {
  size_t idx = (size_t)blockIdx.x * 256 + threadIdx.x;  // S*BL*64
  int di = idx & 63;
  int l = (int)((idx >> 6) & (L_ - 1));
  int b = (int)((idx >> 18) & 3);
  int s = (int)(idx >> 20);
  const float* base = xz + ((size_t)s * BL_ + (size_t)b * L_) * 2 * DI_ + di;
  float acc = cb[s * DI_ + di];
#pragma unroll
  for (int k = 0; k < 4; ++k) {
    int ll = l + k - 3;
    if (ll >= 0) acc += base[(size_t)ll * 2 * DI_] * cw[(s * DI_ + di) * 4 + k];
  }
  xc[((size_t)s * BL_ + (size_t)b * L_ + l) * DI_ + di] = silu_f(acc);
}

// ---------------- kernel 4: x_proj GEMM (BL,64)@(64,48) per split ----------------
__global__ __launch_bounds__(128) void xproj_kernel(
    const float* __restrict__ xc, const float* __restrict__ xpwT,
    float* __restrict__ projp) {
  int widx = blockIdx.x * 4 + (threadIdx.x >> 5);  // S*1024*3 tiles
  int nt = widx % 3;
  int mt = (widx / 3) & 1023;
  int s = widx / 3072;
  const float* A = xc + ((size_t)s * BL_ + mt * 16) * DI_;
  const float* BT = xpwT + (size_t)s * NP_ * DI_ + nt * 16 * DI_;
  v8f acc = {0.f, 0.f, 0.f, 0.f, 0.f, 0.f, 0.f, 0.f};
  gemm_tile_t(A, DI_, BT, DI_, DI_, acc);
  store_tile(projp + ((size_t)s * BL_ + mt * 16) * NP_ + nt * 16, NP_, acc);
}

// ---------------- kernel 5: selective scan + gating ----------------
// 128 single-wave blocks: (seq = s,b) x 8 di-groups of 8. Thread t owns
// di = dgrp*8 + t/4 and 4 states (t&3)*4..+4. Shared proj row broadcast by
// shuffles; next-row loads double-buffered in registers; no LDS, no barriers.
__global__ __launch_bounds__(32) void scan_kernel(
    const float* __restrict__ projp, const float* __restrict__ xc,
    const float* __restrict__ xz, const float* __restrict__ dtw,
    const float* __restrict__ dtb, const float* __restrict__ A_log,
    const float* __restrict__ Dp, float* __restrict__ yg) {
  const int blk = blockIdx.x;
  const int seq = blk >> 3;
  const int dgrp = blk & 7;
  const int s = seq & 3;
  const int b = seq >> 2;
  const int t = threadIdx.x;  // == lane, single wave
  const int di = dgrp * 8 + (t >> 2);
  const int sg = (t & 3) * 4;

  const float a0 = -__expf(A_log[((size_t)s * DI_ + di) * DS_ + sg + 0]);
  const float a1 = -__expf(A_log[((size_t)s * DI_ + di) * DS_ + sg + 1]);
  const float a2 = -__expf(A_log[((size_t)s * DI_ + di) * DS_ + sg + 2]);
  const float a3 = -__expf(A_log[((size_t)s * DI_ + di) * DS_ + sg + 3]);
  const float w0 = dtw[s * 2 * DI_ + di];
  const float w1 = dtw[s * 2 * DI_ + DI_ + di];
  const float db = dtb[s * DI_ + di];
  const float dp = Dp[s * DI_ + di];

  const float* prow = projp + ((size_t)s * BL_ + (size_t)b * L_) * NP_;
  const float* xcb = xc + ((size_t)s * BL_ + (size_t)b * L_) * DI_ + di;
  const float* zb = xz + ((size_t)s * BL_ + (size_t)b * L_) * 2 * DI_ + DI_ + di;
  float* ygb = yg + ((size_t)s * BL_ + (size_t)b * L_) * DI_ + di;

  // preload row 0
  float vp0 = prow[t];
  float vp1 = prow[32 + t];
  float xv = xcb[0];
  float zv = zb[0];
  float h0 = 0.f, h1 = 0.f, h2 = 0.f, h3 = 0.f;

  for (int l = 0; l < L_; ++l) {
    // issue next-step loads before this step's compute (latency overlap)
    const int ln = (l + 1 < L_) ? (l + 1) : l;
    float vp0n = prow[(size_t)ln * NP_ + t];
    float vp1n = prow[(size_t)ln * NP_ + 32 + t];
    float xvn = xcb[(size_t)ln * DI_];
    float zvn = zb[(size_t)ln * 2 * DI_];
    if (t == 0 && l + 16 < L_) {
      __builtin_prefetch(prow + (size_t)(l + 16) * NP_, 0, 1);
      __builtin_prefetch(xcb + (size_t)(l + 16) * DI_, 0, 1);
      __builtin_prefetch(zb + (size_t)(l + 16) * 2 * DI_, 0, 1);
    }

    const float p0 = __shfl(vp0, 0, 32);
    const float p1 = __shfl(vp0, 1, 32);
    float pre = fmaf(p0, w0, fmaf(p1, w1, db));
    const float dt = (pre > 20.f) ? pre : log1pf(__expf(pre));  // softplus
    const float dx = dt * xv;
    const float bv0 = __shfl(vp0, 2 + sg + 0, 32);
    const float bv1 = __shfl(vp0, 2 + sg + 1, 32);
    const float bv2 = __shfl(vp0, 2 + sg + 2, 32);
    const float bv3 = __shfl(vp0, 2 + sg + 3, 32);
    float cv[4];
#pragma unroll
    for (int j = 0; j < 4; ++j) {
      const int col = 18 + sg + j;  // 18..33
      const float lo = __shfl(vp0, col & 31, 32);
      const float hi = __shfl(vp1, (col - 32) & 31, 32);
      cv[j] = (col < 32) ? lo : hi;
    }
    h0 = __expf(dt * a0) * h0 + dx * bv0;
    h1 = __expf(dt * a1) * h1 + dx * bv1;
    h2 = __expf(dt * a2) * h2 + dx * bv2;
    h3 = __expf(dt * a3) * h3 + dx * bv3;
    float acc = h0 * cv[0] + h1 * cv[1] + h2 * cv[2] + h3 * cv[3];
    acc += __shfl_xor(acc, 1, 32);
    acc += __shfl_xor(acc, 2, 32);
    if ((t & 3) == 0)
      ygb[(size_t)l * DI_] = (acc + dp * xv) * silu_f(zv);
    vp0 = vp0n; vp1 = vp1n; xv = xvn; zv = zvn;
  }
}

// ---------------- kernel 6: out_proj -> +skip*xnp -> pvm_proj (+pb) ----------------
__global__ __launch_bounds__(128) void outproj_kernel(
    const float* __restrict__ yg, const float* __restrict__ xnp,
    const float* __restrict__ owT, const float* __restrict__ skip,
    const float* __restrict__ pwT, const float* __restrict__ pb,
    float* __restrict__ ycomb) {
  __shared__ float tmp[4][16 * D_];
  const int w = threadIdx.x >> 5;
  const int widx = blockIdx.x * 4 + w;  // S*1024 tiles
  const int mt = widx & 1023;
  const int s = widx >> 10;
  const int lane = threadIdx.x & 31;
  const int hi = lane >> 4, lr = lane & 15;
  const float sk = skip[s];
  const float* A = yg + ((size_t)s * BL_ + mt * 16) * DI_;
#pragma unroll
  for (int nt = 0; nt < 2; ++nt) {
    v8f acc = {0.f, 0.f, 0.f, 0.f, 0.f, 0.f, 0.f, 0.f};
    gemm_tile_t(A, DI_, owT + (size_t)s * D_ * DI_ + nt * 16 * DI_, DI_, DI_,
                acc);
#pragma unroll
    for (int r = 0; r < 8; ++r) {
      const int m = r + 8 * hi;
      acc[r] += sk * xnp[((size_t)s * BL_ + mt * 16 + m) * D_ + nt * 16 + lr];
      tmp[w][m * D_ + nt * 16 + lr] = acc[r];
    }
  }
  __syncthreads();
#pragma unroll
  for (int nt = 0; nt < 2; ++nt) {
    v8f acc = {0.f, 0.f, 0.f, 0.f, 0.f, 0.f, 0.f, 0.f};
    gemm_tile_t(&tmp[w][0], D_, pwT + (size_t)s * D_ * D_ + nt * 16 * D_, D_,
                D_, acc);
#pragma unroll
    for (int r = 0; r < 8; ++r) {
      const int m = r + 8 * hi;
      ycomb[((size_t)mt * 16 + m) * C_ + s * D_ + nt * 16 + lr] =
          acc[r] + pb[s * D_ + nt * 16 + lr];
    }
  }
}

// ---------------- kernel 7: final 128x128 GEMM + BN + GELU, -> NCHW ----------------
__global__ __launch_bounds__(128) void final_kernel(
    const float* __restrict__ ycomb, const float* __restrict__ pw,
    const float* __restrict__ pbias, const float* __restrict__ g,
    const float* __restrict__ bb, const float* __restrict__ mean,
    const float* __restrict__ var, float* __restrict__ out) {
  const int widx = blockIdx.x * 4 + (threadIdx.x >> 5);  // 1024*8 tiles
  const int nt = widx & 7;
  const int mt = widx >> 3;
  const int lane = threadIdx.x & 31;
  const int hi = lane >> 4, lr = lane & 15;
  const int n0 = nt * 16, m0 = mt * 16;
  const int o = n0 + lr;  // output channel
  v8f acc = {0.f, 0.f, 0.f, 0.f, 0.f, 0.f, 0.f, 0.f};
  // proj_w[o][c] is already B-transposed layout: BT row = o, K-contiguous
  gemm_tile_t(ycomb + (size_t)m0 * C_, C_, pw + (size_t)n0 * C_, C_, C_, acc);
  const float scale = g[o] * rsqrtf(var[o] + EPS_);
  const float bias = pbias[o];
#pragma unroll
  for (int r = 0; r < 8; ++r) {
    const int p = m0 + r + 8 * hi;
    const int bidx = p >> 12;
    const int l = p & (L_ - 1);
    float v = acc[r] + bias;
    v = (v - mean[o]) * scale + bb[o];
    v = 0.5f * v * (1.f + erff(v * 0.70710678118654752f));  // exact GELU
    out[((size_t)bidx * C_ + o) * L_ + l] = v;
  }
}

// ---------------- launch ----------------
extern "C" void kernel_launch(void* const* d_in, const int* in_sizes, int n_in,
                              void* d_out, int out_size, void* d_ws,
                              size_t ws_size, hipStream_t stream) {
  (void)in_sizes; (void)n_in; (void)out_size; (void)ws_size;
  const float* x        = (const float*)d_in[0];
  const float* ln_g     = (const float*)d_in[1];
  const float* ln_b     = (const float*)d_in[2];
  const float* pvm_ln_g = (const float*)d_in[3];
  const float* pvm_ln_b = (const float*)d_in[4];
  const float* in_w     = (const float*)d_in[5];
  const float* conv_w   = (const float*)d_in[6];
  const float* conv_b   = (const float*)d_in[7];
  const float* x_proj_w = (const float*)d_in[8];
  const float* dt_w     = (const float*)d_in[9];
  const float* dt_b     = (const float*)d_in[10];
  const float* A_log    = (const float*)d_in[11];
  const float* D_param  = (const float*)d_in[12];
  const float* out_w    = (const float*)d_in[13];
  const float* skip     = (const float*)d_in[14];
  const float* pvm_w    = (const float*)d_in[15];
  const float* pvm_b    = (const float*)d_in[16];
  const float* proj_w   = (const float*)d_in[17];
  const float* proj_b   = (const float*)d_in[18];
  const float* bn_g     = (const float*)d_in[19];
  const float* bn_b     = (const float*)d_in[20];
  const float* bn_mean  = (const float*)d_in[21];
  const float* bn_var   = (const float*)d_in[22];
  float* out = (float*)d_out;
  float* ws = (float*)d_ws;

  float* xnp   = ws + OFF_XNP;
  float* xz    = ws + OFF_XZ;
  float* xc    = ws + OFF_XC;
  float* projp = ws + OFF_PR;
  float* yg    = ws + OFF_YG;
  float* ycomb = ws + OFF_YC;
  float* in_wT = ws + OFF_INWT;
  float* xpwT  = ws + OFF_XPWT;
  float* owT   = ws + OFF_OWT;
  float* pwT   = ws + OFF_PWT;

  constexpr int PREP_N =
      S_ * C_ * D_ + S_ * NP_ * DI_ + S_ * D_ * DI_ + S_ * D_ * D_;
  prep_kernel<<<(PREP_N + 255) / 256, 256, 0, stream>>>(
      in_w, x_proj_w, out_w, pvm_w, in_wT, xpwT, owT, pwT);
  ln_kernel<<<BL_ / 8, 256, 0, stream>>>(x, ln_g, ln_b, pvm_ln_g, pvm_ln_b,
                                         xnp);
  inproj_kernel<<<(S_ * (BL_ / 16) * 8) / 4, 128, 0, stream>>>(xnp, in_wT, xz);
  conv_kernel<<<(S_ * BL_ * DI_) / 256, 256, 0, stream>>>(xz, conv_w, conv_b,
                                                          xc);
  xproj_kernel<<<(S_ * (BL_ / 16) * 3) / 4, 128, 0, stream>>>(xc, xpwT, projp);
  scan_kernel<<<S_ * B_ * 8, 32, 0, stream>>>(projp, xc, xz, dt_w, dt_b, A_log,
                                              D_param, yg);
  outproj_kernel<<<(S_ * (BL_ / 16)) / 4, 128, 0, stream>>>(
      yg, xnp, owT, skip, pwT, pvm_b, ycomb);
  final_kernel<<<((BL_ / 16) * 8) / 4, 128, 0, stream>>>(
      ycomb, proj_w, proj_b, bn_g, bn_b, bn_mean, bn_var, out);
}